// VerticalMedian_88923002896727
// MI455X (gfx1250) — compile-verified
//
#include <hip/hip_runtime.h>

// Vertical 3-tap median filter with mask select.
// Memory-bound streaming kernel: B128 loads/stores, v_med3_f32, NT stores,
// global_prefetch for L2 warming. ~232 MB total traffic -> ~10 us at 23.3 TB/s.

typedef float v4f __attribute__((ext_vector_type(4)));

constexpr int Hc   = 2160;
constexpr int Wc   = 3840;
constexpr int Cc   = 3;
constexpr int ROWF = Wc * Cc;      // 11520 floats per image row (divisible by 4)
constexpr int ROWV = ROWF / 4;     // 2880 float4-vectors per row
constexpr int NVEC = Hc * ROWV;    // 6,220,800 vectors total

__global__ __launch_bounds__(256)
void VerticalMedian_kernel(const float* __restrict__ img,
                           const int*   __restrict__ mask,
                           float*       __restrict__ out)
{
    const int v = blockIdx.x * 256 + threadIdx.x;
    if (v >= NVEC) return;

    const int h  = v / ROWV;          // image row
    const int rv = v - h * ROWV;      // vector index within row
    const int r  = rv * 4;            // float index within row

    const v4f* __restrict__ imgv = (const v4f*)img;
    v4f*       __restrict__ outv = (v4f*)out;

    // Symmetric pad == clamp: row -1 -> 0, row H -> H-1 (med(x,x,c)=x at edges).
    const int up = (h > 0)      ? v - ROWV : v;
    const int dn = (h < Hc - 1) ? v + ROWV : v;

    const v4f b = imgv[v];            // center row (also the mask==0 passthrough)
    const v4f a = imgv[up];
    const v4f c = imgv[dn];

    // Warm L2 two rows below for the blocks that will consume row h+2.
    // global_prefetch: no VGPR return, no LOADcnt -- essentially free.
    if (h + 2 < Hc)
        __builtin_prefetch((const void*)(imgv + v + 2 * ROWV), 0, 1);

    // Per-component mask: 4 channel slots span exactly 2 pixels.
    const int p0 = r / 3;             // first pixel covered by this vector
    const int c0 = r - 3 * p0;        // starting channel (0..2)
    const int mbase = h * Wc + p0;
    const int m0 = mask[mbase];
    const int m1 = mask[mbase + 1];   // p0 <= W-2 for any aligned vector

    // median-of-3 == v_med3_f32 (finite inputs, matches max(min..)/min(max..))
    v4f med;
    med.x = __builtin_amdgcn_fmed3f(a.x, b.x, c.x);
    med.y = __builtin_amdgcn_fmed3f(a.y, b.y, c.y);
    med.z = __builtin_amdgcn_fmed3f(a.z, b.z, c.z);
    med.w = __builtin_amdgcn_fmed3f(a.w, b.w, c.w);

    // Component k belongs to pixel p0 + ((c0+k) >= 3).
    v4f o;
    o.x = (((c0 + 0 >= 3) ? m1 : m0) == 1) ? med.x : b.x;
    o.y = (((c0 + 1 >= 3) ? m1 : m0) == 1) ? med.y : b.y;
    o.z = (((c0 + 2 >= 3) ? m1 : m0) == 1) ? med.z : b.z;
    o.w = (((c0 + 3 >= 3) ? m1 : m0) == 1) ? med.w : b.w;

    // Output is write-once: non-temporal store keeps L2 free for img row reuse.
    __builtin_nontemporal_store(o, outv + v);
}

extern "C" void kernel_launch(void* const* d_in, const int* in_sizes, int n_in,
                              void* d_out, int out_size, void* d_ws, size_t ws_size,
                              hipStream_t stream)
{
    const float* img  = (const float*)d_in[0];
    const int*   mask = (const int*)d_in[1];
    float*       out  = (float*)d_out;

    const int blocks = (NVEC + 255) / 256;   // 24300 blocks of 8 wave32s
    VerticalMedian_kernel<<<blocks, 256, 0, stream>>>(img, mask, out);
}